// GNNDecoder_v2_50955491999989
// MI455X (gfx1250) — compile-verified
//
#include <hip/hip_runtime.h>

#define N_NODES 50000
#define N_EDGES 800000
#define DIM     128

typedef __attribute__((ext_vector_type(2))) float v2f;
typedef __attribute__((ext_vector_type(8))) float v8f;

// ---------------------------------------------------------------------------
// Zero the aggregation buffer (workspace is poisoned, must re-zero every call)
// ---------------------------------------------------------------------------
__global__ __launch_bounds__(256) void zero_f4(float4* __restrict__ p, int n4) {
    int i = blockIdx.x * blockDim.x + threadIdx.x;
    if (i < n4) p[i] = make_float4(0.f, 0.f, 0.f, 0.f);
}

// ---------------------------------------------------------------------------
// Edge phase: aggr[dst] += x[src] + ee1[attr0] + ee2[attr1]
// One wave (32 lanes) per edge; each lane owns 4 contiguous floats (128 = 32*4).
// x gathers are 512B/edge, L2-resident (x fits in 192MB L2). Aggregation via
// global_atomic_add_f32 (4 per lane).
// ---------------------------------------------------------------------------
__global__ __launch_bounds__(256) void scatter_add(
    const float4* __restrict__ X,      // [N, 32] as float4
    const int*    __restrict__ srcIdx, // [E]
    const int*    __restrict__ dstIdx, // [E]
    const int*    __restrict__ attr,   // [E, 2]
    const float4* __restrict__ EE1,    // [6, 32] as float4
    const float4* __restrict__ EE2,    // [3, 32] as float4
    float*        __restrict__ aggr,   // [N, 128]
    int E)
{
    int t = blockIdx.x * blockDim.x + threadIdx.x;
    int e = t >> 5;
    int c = t & 31;
    if (e >= E) return;

    int s  = srcIdx[e];
    int d  = dstIdx[e];
    int a0 = attr[2 * e + 0];
    int a1 = attr[2 * e + 1];

    float4 xv = X[(size_t)s * 32 + c];
    float4 e1 = EE1[a0 * 32 + c];
    float4 e2 = EE2[a1 * 32 + c];

    float* out = aggr + (size_t)d * DIM + c * 4;
    atomicAdd(out + 0, xv.x + e1.x + e2.x);
    atomicAdd(out + 1, xv.y + e1.y + e2.y);
    atomicAdd(out + 2, xv.z + e1.z + e2.z);
    atomicAdd(out + 3, xv.w + e1.w + e2.w);
}

// ---------------------------------------------------------------------------
// Out = act(A[M,K] @ W[K,NC] + bias[NC]) using V_WMMA_F32_16X16X4_F32.
// One 16x16 output tile per wave. fp32 A/B/C -> matches reference numerics.
//   A layout : lane(0-15)=row M, half=lane>>4 selects K pair -> float2 load
//   B layout : lane&15 = col N, half selects K pair -> two strided loads
//   C layout : c[r] = C[r + 8*half][lane&15]
// ---------------------------------------------------------------------------
__global__ __launch_bounds__(256) void gemm_bias_act(
    const float* __restrict__ A,    // [M, K]
    const float* __restrict__ W,    // [K, NC]
    const float* __restrict__ bias, // [NC]
    float*       __restrict__ Out,  // [M, NC]
    int M, int K, int NC, int relu)
{
    int wid  = (blockIdx.x * blockDim.x + threadIdx.x) >> 5;
    int lane = threadIdx.x & 31;

    int tilesN     = NC >> 4;
    int totalTiles = (M >> 4) * tilesN;
    if (wid >= totalTiles) return;          // wave-uniform exit (EXEC all-1 inside)

    int tm   = wid / tilesN;
    int tn   = wid - tm * tilesN;
    int half = lane >> 4;                   // 0 or 1
    int l15  = lane & 15;

    const float* Arow = A + (size_t)(tm * 16 + l15) * K;   // row m = l15
    const float* Bcol = W + tn * 16 + l15;                 // col n = l15

    v8f acc = {};
    for (int kb = 0; kb < K; kb += 4) {
        int ka = kb + 2 * half;
        float2 af = *(const float2*)(Arow + ka);           // A[m][ka], A[m][ka+1]
        v2f a; a.x = af.x; a.y = af.y;
        v2f b;
        b.x = Bcol[(size_t)(ka + 0) * NC];                 // B[ka][n]
        b.y = Bcol[(size_t)(ka + 1) * NC];                 // B[ka+1][n]
        acc = __builtin_amdgcn_wmma_f32_16x16x4_f32(
            /*neg_a=*/false, a, /*neg_b=*/false, b,
            /*c_mod=*/(short)0, acc, /*reuse_a=*/false, /*reuse_b=*/false);
    }

    int   col = tn * 16 + l15;
    float bv  = bias[col];
    float* outBase = Out + (size_t)(tm * 16 + 8 * half) * NC + col;
#pragma unroll
    for (int r = 0; r < 8; ++r) {
        float v = acc[r] + bv;
        if (relu) v = fmaxf(v, 0.f);
        outBase[(size_t)r * NC] = v;
    }
}

// ---------------------------------------------------------------------------
// Driver: 2 GIN layers
//   aggr = scatter(x);  h = relu(aggr@W1+b1);  y = h@W2+b2;  z = relu(y)  (L0)
//   aggr = scatter(z);  h = relu(aggr@W1+b1);  out = h@W2+b2             (L1)
// ---------------------------------------------------------------------------
extern "C" void kernel_launch(void* const* d_in, const int* in_sizes, int n_in,
                              void* d_out, int out_size, void* d_ws, size_t ws_size,
                              hipStream_t stream) {
    const float* x     = (const float*)d_in[0];
    const int*   eidx  = (const int*)  d_in[1];
    const int*   eattr = (const int*)  d_in[2];
    const float* ee1_0 = (const float*)d_in[3];
    const float* ee2_0 = (const float*)d_in[4];
    const float* W1_0  = (const float*)d_in[5];
    const float* b1_0  = (const float*)d_in[6];
    const float* W2_0  = (const float*)d_in[7];
    const float* b2_0  = (const float*)d_in[8];
    const float* ee1_1 = (const float*)d_in[9];
    const float* ee2_1 = (const float*)d_in[10];
    const float* W1_1  = (const float*)d_in[11];
    const float* b1_1  = (const float*)d_in[12];
    const float* W2_1  = (const float*)d_in[13];
    const float* b2_1  = (const float*)d_in[14];

    const int E = in_sizes[1] / 2;          // 800000
    const int N = in_sizes[0] / DIM;        // 50000
    const int* srcI = eidx;
    const int* dstI = eidx + E;

    float* aggr = (float*)d_ws;                         // [N, 128]
    float* h    = aggr + (size_t)N * DIM;               // [N, 256]
    float* z    = h    + (size_t)N * 2 * DIM;           // [N, 128]

    const dim3 blk(256);
    const int  n4 = N * DIM / 4;
    const dim3 zg((n4 + 255) / 256);
    const dim3 sg(((size_t)E * 32 + 255) / 256);
    const int  t1 = (N / 16) * (2 * DIM / 16);          // GEMM1 tiles (NC=256)
    const int  t2 = (N / 16) * (DIM / 16);              // GEMM2 tiles (NC=128)
    const dim3 g1((t1 * 32 + 255) / 256);
    const dim3 g2((t2 * 32 + 255) / 256);

    // ---- layer 0 ----
    zero_f4<<<zg, blk, 0, stream>>>((float4*)aggr, n4);
    scatter_add<<<sg, blk, 0, stream>>>((const float4*)x, srcI, dstI, eattr,
                                        (const float4*)ee1_0, (const float4*)ee2_0,
                                        aggr, E);
    gemm_bias_act<<<g1, blk, 0, stream>>>(aggr, W1_0, b1_0, h, N, DIM,     2 * DIM, 1);
    gemm_bias_act<<<g2, blk, 0, stream>>>(h,    W2_0, b2_0, z, N, 2 * DIM, DIM,     1);

    // ---- layer 1 ----
    zero_f4<<<zg, blk, 0, stream>>>((float4*)aggr, n4);
    scatter_add<<<sg, blk, 0, stream>>>((const float4*)z, srcI, dstI, eattr,
                                        (const float4*)ee1_1, (const float4*)ee2_1,
                                        aggr, E);
    gemm_bias_act<<<g1, blk, 0, stream>>>(aggr, W1_1, b1_1, h,            N, DIM,     2 * DIM, 1);
    gemm_bias_act<<<g2, blk, 0, stream>>>(h,    W2_1, b2_1, (float*)d_out, N, 2 * DIM, DIM,    0);
}